// RSRConv_39745627357803
// MI455X (gfx1250) — compile-verified
//
#include <hip/hip_runtime.h>
#include <hip/hip_bf16.h>
#include <math.h>

typedef __attribute__((ext_vector_type(2))) float v2f;
typedef __attribute__((ext_vector_type(8))) float v8f;

#define FDIM 64

// ---- monotone float<->int ordering for atomicMax on f32 -------------------
__device__ __forceinline__ int f32_to_ordered(float f) {
    int i = __float_as_int(f);
    return (i >= 0) ? i : (i ^ 0x7FFFFFFF);
}
__device__ __forceinline__ float ordered_to_f32(int i) {
    return __int_as_float((i >= 0) ? i : (i ^ 0x7FFFFFFF));
}

// ---- k0: zero output, init segment-max / denom ----------------------------
__global__ void k_init(float* __restrict__ out, int* __restrict__ emax,
                       float* __restrict__ denom, int n_out, int n_nodes) {
    int idx = blockIdx.x * blockDim.x + threadIdx.x;
    if (idx < n_out) out[idx] = 0.0f;
    if (idx < n_nodes) {
        emax[idx] = (int)0x80000000;   // ordered-int -inf
        denom[idx] = 0.0f;
    }
}

// ---- k1: el/er via V_WMMA_F32_16X16X4_F32 ---------------------------------
// One wave computes el,er for 16 nodes. D = A(16x4) x B(4x16) accumulated
// over 16 K-chunks of 4. A row0 = head_w chunk, row1 = tail_w chunk, rest 0.
// B[k][n] = nfeat[node_base+n][kc+k]. D[0][n]=el, D[1][n]=er land in
// acc[0]/acc[1] of lanes 0..15 (32-bit C/D layout: VGPR r = row M=r).
// A-fragment built branch-free (unconditional b64 weight loads + cndmask)
// so no EXEC save/restore appears between loads and WMMA.
__global__ void k_scores(const float* __restrict__ nfeat,
                         const float* __restrict__ head_w,
                         const float* __restrict__ tail_w,
                         const float* __restrict__ head_b,
                         const float* __restrict__ tail_b,
                         float* __restrict__ el, float* __restrict__ er,
                         int n_nodes) {
    const int lane = threadIdx.x & 31;
    const int wave = threadIdx.x >> 5;
    const int node_base = (blockIdx.x * (blockDim.x >> 5) + wave) * 16;
    if (node_base >= n_nodes) return;            // wave-uniform exit

    const int mrow = lane & 15;                  // A row / B column (node)
    const int kofs = (lane >> 4) << 1;           // K sub-offset: 0 or 2
    const bool is_h = (mrow == 0);               // lane supplies head_w row
    const bool is_t = (mrow == 1);               // lane supplies tail_w row

    int node = node_base + mrow;
    if (node >= n_nodes) node = n_nodes - 1;     // clamp loads, mask stores
    const float* nrow = nfeat + (size_t)node * FDIM;

    v8f acc = {};
#pragma unroll
    for (int kc = 0; kc < FDIM; kc += 4) {
        v2f h = *(const v2f*)(head_w + kc + kofs);   // unconditional b64 loads
        v2f t = *(const v2f*)(tail_w + kc + kofs);
        v2f a;
        a[0] = is_h ? h[0] : (is_t ? t[0] : 0.0f);   // branch-free selects
        a[1] = is_h ? h[1] : (is_t ? t[1] : 0.0f);
        v2f b = *(const v2f*)(nrow + kc + kofs);     // B[kofs..+1][node col]
        acc = __builtin_amdgcn_wmma_f32_16x16x4_f32(
            /*neg_a=*/false, a, /*neg_b=*/false, b,
            /*c_mod=*/(short)0, acc, /*reuse_a=*/false, /*reuse_b=*/false);
    }

    if (lane < 16) {
        int nd = node_base + lane;               // D column N == lane
        if (nd < n_nodes) {
            el[nd] = acc[0] + head_b[0];         // M=0 row
            er[nd] = acc[1] + tail_b[0];         // M=1 row
        }
    }
}

// ---- k2: per-edge score + segment max -------------------------------------
__global__ void k_edge_score(const int* __restrict__ src, const int* __restrict__ dst,
                             const int* __restrict__ etype,
                             const float* __restrict__ el, const float* __restrict__ er,
                             const float* __restrict__ rel_w,
                             float* __restrict__ e_buf, int* __restrict__ emax,
                             int n_edges) {
    int i = blockIdx.x * blockDim.x + threadIdx.x;
    if (i >= n_edges) return;
    int s  = __builtin_nontemporal_load(src + i);    // streaming edge lists: TH=NT
    int d  = __builtin_nontemporal_load(dst + i);
    int et = __builtin_nontemporal_load(etype + i);
    float e = el[s] + er[d] + rel_w[et];
    __builtin_nontemporal_store(e, e_buf + i);
    atomicMax(&emax[d], f32_to_ordered(e));          // global_atomic_max_i32
}

// ---- k3: exp(e - max) + segment sum ---------------------------------------
__global__ void k_edge_exp(const int* __restrict__ dst,
                           const int* __restrict__ emax,
                           float* __restrict__ e_buf,    // in: e, out: exp
                           float* __restrict__ denom, int n_edges) {
    int i = blockIdx.x * blockDim.x + threadIdx.x;
    if (i >= n_edges) return;
    int d = __builtin_nontemporal_load(dst + i);
    float ex = __expf(e_buf[i] - ordered_to_f32(emax[d]));
    e_buf[i] = ex;
    atomicAdd(&denom[d], ex);                        // global_atomic_add_f32
}

// ---- k4: out[dst] += a * nfeat[src]; 16 lanes/edge, float4/lane -----------
__global__ void k_aggregate(const int* __restrict__ src, const int* __restrict__ dst,
                            const float* __restrict__ nfeat,
                            const float* __restrict__ e_buf,   // exp values
                            const float* __restrict__ denom,
                            float* __restrict__ out, int n_edges) {
    int t = blockIdx.x * blockDim.x + threadIdx.x;
    int edge = t >> 4;
    if (edge >= n_edges) return;
    int f4 = (t & 15) << 2;
    int s = src[edge], d = dst[edge];
    float a = e_buf[edge] / denom[d];
    const float4 v = *(const float4*)(nfeat + (size_t)s * FDIM + f4);  // b128 gather
    float* o = out + (size_t)d * FDIM + f4;
    atomicAdd(o + 0, a * v.x);
    atomicAdd(o + 1, a * v.y);
    atomicAdd(o + 2, a * v.z);
    atomicAdd(o + 3, a * v.w);
}

// ---- launch ---------------------------------------------------------------
extern "C" void kernel_launch(void* const* d_in, const int* in_sizes, int n_in,
                              void* d_out, int out_size, void* d_ws, size_t ws_size,
                              hipStream_t stream) {
    const float* nfeat  = (const float*)d_in[0];
    const int*   src    = (const int*)d_in[1];
    const int*   dst    = (const int*)d_in[2];
    const int*   etype  = (const int*)d_in[3];
    const float* head_w = (const float*)d_in[4];
    const float* head_b = (const float*)d_in[5];
    const float* tail_w = (const float*)d_in[6];
    const float* tail_b = (const float*)d_in[7];
    const float* rel_w  = (const float*)d_in[8];
    float* out = (float*)d_out;

    const int n_nodes = in_sizes[0] / FDIM;   // 50000
    const int n_edges = in_sizes[1];          // 800000

    // workspace layout (floats): el | er | emax(int) | denom | e_buf
    float* ws    = (float*)d_ws;
    float* el    = ws;
    float* er    = ws + (size_t)n_nodes;
    int*   emax  = (int*)(ws + 2 * (size_t)n_nodes);
    float* denom = ws + 3 * (size_t)n_nodes;
    float* e_buf = ws + 4 * (size_t)n_nodes;

    const int n_out = n_nodes * FDIM;
    k_init<<<(n_out + 255) / 256, 256, 0, stream>>>(out, emax, denom, n_out, n_nodes);

    {
        int waves  = (n_nodes + 15) / 16;     // one wave per 16 nodes
        int blocks = (waves + 7) / 8;         // 8 waves / 256-thread block
        k_scores<<<blocks, 256, 0, stream>>>(nfeat, head_w, tail_w, head_b, tail_b,
                                             el, er, n_nodes);
    }

    k_edge_score<<<(n_edges + 255) / 256, 256, 0, stream>>>(
        src, dst, etype, el, er, rel_w, e_buf, emax, n_edges);

    k_edge_exp<<<(n_edges + 255) / 256, 256, 0, stream>>>(
        dst, emax, e_buf, denom, n_edges);

    {
        long long threads = (long long)n_edges * 16;
        int blocks = (int)((threads + 255) / 256);
        k_aggregate<<<blocks, 256, 0, stream>>>(src, dst, nfeat, e_buf, denom,
                                                out, n_edges);
    }
}